// CannyFilter_6511170421175
// MI455X (gfx1250) — compile-verified
//
#include <hip/hip_runtime.h>
#include <math.h>

#define IMG_H 1024
#define IMG_W 1024
#define PH 1044          // padded (reflect PAD=10) field
#define PW 1044
#define NB 8
#define NC 3

typedef __attribute__((ext_vector_type(2))) float v2f;
typedef __attribute__((ext_vector_type(8))) float v8f;

#if defined(__has_builtin)
#if __has_builtin(__builtin_amdgcn_wmma_f32_16x16x4_f32)
#define HAVE_WMMA_F32X4 1
#endif
#endif
#ifndef HAVE_WMMA_F32X4
#define HAVE_WMMA_F32X4 0
#endif

// -------------------------------------------------------------------------
// K1: depthwise gaussian blur on reflect-padded field + channel sum.
// One wave (32 threads) per 16x16 output tile; 3x3 conv done as three banded
// 16x18 x 18x16 matmuls (one per kernel column) -> 15 chained f32 WMMAs.
// -------------------------------------------------------------------------
__global__ __launch_bounds__(32)
void canny_blur_sum_kernel(const float* __restrict__ img,
                           const float* __restrict__ gw,   // [3][3]
                           float* __restrict__ blurred,    // [NB][NC][PH][PW]
                           float* __restrict__ summed) {   // [NB][PH][PW]
  __shared__ float tile[20][20];
  const int l   = threadIdx.x;       // 0..31
  const int n   = l & 15;            // column within tile / row M for A
  const int hi  = l >> 4;            // lane-half
  const int tx0 = blockIdx.x * 16;
  const int ty0 = blockIdx.y * 16;
  const int b   = blockIdx.z;

  float w[9];
#pragma unroll
  for (int t = 0; t < 9; ++t) w[t] = gw[t];

  // zero whole LDS tile once (rows/cols 18,19 stay 0 -> safe zero band)
  for (int t = l; t < 20 * 20; t += 32) tile[t / 20][t % 20] = 0.0f;
  __syncthreads();

  v8f sumacc = {0.f, 0.f, 0.f, 0.f, 0.f, 0.f, 0.f, 0.f};

  for (int c = 0; c < NC; ++c) {
    const float* src = img + ((size_t)b * NC + c) * (size_t)IMG_H * IMG_W;
    // Load 18x18 tile of P (reflect-padded img inside [0,PH), zero outside).
    for (int t = l; t < 18 * 18; t += 32) {
      const int r = t / 18, q = t % 18;
      const int gi = ty0 - 1 + r;
      const int gj = tx0 - 1 + q;
      float v = 0.0f;
      if (gi >= 0 && gi < PH && gj >= 0 && gj < PW) {
        int ri = gi - 10; ri = ri < 0 ? -ri : (ri > IMG_H - 1 ? 2 * (IMG_H - 1) - ri : ri);
        int rj = gj - 10; rj = rj < 0 ? -rj : (rj > IMG_W - 1 ? 2 * (IMG_W - 1) - rj : rj);
        v = src[(size_t)ri * IMG_W + rj];
      }
      tile[r][q] = v;
    }
    __syncthreads();

    v8f acc = {0.f, 0.f, 0.f, 0.f, 0.f, 0.f, 0.f, 0.f};
#if HAVE_WMMA_F32X4
    // OUT = sum_dx  Band(w[:,dx])[16x18] * P[:, dx-shifted][18x16]
    // A 16x4 layout: lanes 0-15 hold (K0,K1) for M=lane, lanes 16-31 (K2,K3).
    // B 4x16 layout: VGPR v, lane-half hi -> row K = v + 2*hi, col N = lane%16.
#pragma unroll
    for (int c4 = 0; c4 < 5; ++c4) {
      const int r0 = 4 * c4 + 2 * hi;   // first K row this lane touches
#pragma unroll
      for (int dx = 0; dx < 3; ++dx) {
        const int d0 = r0 - n;          // band offset (r - m)
        const int d1 = d0 + 1;
        const float a0 = (d0 >= 0 && d0 <= 2) ? w[d0 * 3 + dx] : 0.0f;
        const float a1 = (d1 >= 0 && d1 <= 2) ? w[d1 * 3 + dx] : 0.0f;
        v2f A = {a0, a1};
        v2f Bv = {tile[r0][n + dx], tile[r0 + 1][n + dx]};  // rows 18/19 are zero
        acc = __builtin_amdgcn_wmma_f32_16x16x4_f32(
            /*neg_a=*/false, A, /*neg_b=*/false, Bv,
            /*c_mod=*/(short)0, acc, /*reuse_a=*/false, /*reuse_b=*/false);
      }
    }
#else
    // VALU fallback: identical D layout (lane = col n, rows r+8*hi).
#pragma unroll
    for (int r = 0; r < 8; ++r) {
      const int y = r + 8 * hi;
      float s = 0.0f;
#pragma unroll
      for (int dy = 0; dy < 3; ++dy)
#pragma unroll
        for (int dx = 0; dx < 3; ++dx) s += w[dy * 3 + dx] * tile[y + dy][n + dx];
      acc[r] = s;
    }
#endif

    float* dstB = blurred + ((size_t)b * NC + c) * (size_t)PH * PW;
#pragma unroll
    for (int r = 0; r < 8; ++r) {
      const int oy = ty0 + r + 8 * hi;
      const int ox = tx0 + n;
      if (oy < PH && ox < PW) dstB[(size_t)oy * PW + ox] = acc[r];
      sumacc[r] += acc[r];
    }
    __syncthreads();   // tile reused next channel
  }

  float* dstS = summed + (size_t)b * PH * PW;
#pragma unroll
  for (int r = 0; r < 8; ++r) {
    const int oy = ty0 + r + 8 * hi;
    const int ox = tx0 + n;
    if (oy < PH && ox < PW) dstS[(size_t)oy * PW + ox] = sumacc[r];
  }
}

// -------------------------------------------------------------------------
// K2: Sobel x/y on summed (cropped interior -> no padding issues), /C,
//     magnitude, orientation rounded to 45-degree bins (RNE like jnp.round).
// -------------------------------------------------------------------------
__global__ __launch_bounds__(256)
void canny_sobel_kernel(const float* __restrict__ summed,
                        const float* __restrict__ sxw,
                        const float* __restrict__ syw,
                        float* __restrict__ gx_out, float* __restrict__ gy_out,
                        float* __restrict__ mag_out, float* __restrict__ ori_out) {
  const size_t idx = (size_t)blockIdx.x * 256 + threadIdx.x;
  const size_t npix = (size_t)NB * IMG_H * IMG_W;
  if (idx >= npix) return;
  const int j = (int)(idx % IMG_W);
  const size_t t = idx / IMG_W;
  const int i = (int)(t % IMG_H);
  const int b = (int)(t / IMG_H);
  const float* s = summed + (size_t)b * PH * PW;

  __builtin_prefetch(&s[(size_t)(i + 13) * PW + (j + 9)], 0, 1);

  float gx = 0.0f, gy = 0.0f;
#pragma unroll
  for (int dy = 0; dy < 3; ++dy)
#pragma unroll
    for (int dx = 0; dx < 3; ++dx) {
      const float v = s[(size_t)(i + 9 + dy) * PW + (j + 9 + dx)];
      gx += sxw[dy * 3 + dx] * v;
      gy += syw[dy * 3 + dx] * v;
    }
  gx *= (1.0f / 3.0f);
  gy *= (1.0f / 3.0f);
  const float mag = sqrtf(gx * gx + gy * gy);
  float ori = atanf(gy / gx) * (360.0f / 3.14159265358979323846f) + 180.0f;
  ori = rintf(ori * (1.0f / 45.0f)) * 45.0f;   // round-half-even, like jnp.round
  gx_out[idx] = gx;
  gy_out[idx] = gy;
  mag_out[idx] = mag;
  ori_out[idx] = ori;
}

// -------------------------------------------------------------------------
// K3: NMS + double threshold. directional[i] = mag(p) - mag(p + delta_i)
// (zero-padded), so min(dir_pos,dir_neg)>0  <=>  mag > both neighbors.
// -------------------------------------------------------------------------
__global__ __launch_bounds__(256)
void canny_nms_kernel(const float* __restrict__ mag_all,
                      const float* __restrict__ ori_all,
                      float* __restrict__ stage_all) {
  const size_t idx = (size_t)blockIdx.x * 256 + threadIdx.x;
  const size_t npix = (size_t)NB * IMG_H * IMG_W;
  if (idx >= npix) return;
  const int j = (int)(idx % IMG_W);
  const size_t t = idx / IMG_W;
  const int i = (int)(t % IMG_H);
  const int b = (int)(t / IMG_H);
  const float* mag = mag_all + (size_t)b * IMG_H * IMG_W;

  const float m = mag[(size_t)i * IMG_W + j];
  const float ori = ori_all[idx];
  float te = m;
  if (ori == ori) {  // NaN orientation matches no pair -> pixel kept
    const int iq = __float2int_rn(ori * (1.0f / 45.0f)) & 7;
    const int pair = iq & 3;
    // pair -> (dy,dx): 0:(0,1) 1:(-1,1) 2:(-1,0) 3:(-1,-1)
    const int dy = (pair == 0) ? 0 : -1;
    const int dx = (pair <= 1) ? 1 : ((pair == 2) ? 0 : -1);
    float m1 = 0.0f, m2 = 0.0f;
    const int i1 = i + dy, j1 = j + dx, i2 = i - dy, j2 = j - dx;
    if ((unsigned)i1 < IMG_H && (unsigned)j1 < IMG_W) m1 = mag[(size_t)i1 * IMG_W + j1];
    if ((unsigned)i2 < IMG_H && (unsigned)j2 < IMG_W) m2 = mag[(size_t)i2 * IMG_W + j2];
    const bool is_max = (m > m1) && (m > m2);
    if (!is_max) te = 0.0f;
  }
  const float stage = (te > 0.05f ? 0.5f : 0.0f) + (te > 0.2f ? 0.5f : 0.0f);
  stage_all[idx] = stage;
}

// -------------------------------------------------------------------------
// K4: hysteresis: weak (==0.5) pixels promoted when 1.25-weighted 3x3 sum > 1.
// -------------------------------------------------------------------------
__global__ __launch_bounds__(256)
void canny_hyst_kernel(const float* __restrict__ stage_all,
                       const float* __restrict__ hw,
                       float* __restrict__ out) {
  const size_t idx = (size_t)blockIdx.x * 256 + threadIdx.x;
  const size_t npix = (size_t)NB * IMG_H * IMG_W;
  if (idx >= npix) return;
  const int j = (int)(idx % IMG_W);
  const size_t t = idx / IMG_W;
  const int i = (int)(t % IMG_H);
  const int b = (int)(t / IMG_H);
  const float* st = stage_all + (size_t)b * IMG_H * IMG_W;

  const float s = st[(size_t)i * IMG_W + j];
  float sum = 0.0f;
#pragma unroll
  for (int dy = 0; dy < 3; ++dy)
#pragma unroll
    for (int dx = 0; dx < 3; ++dx) {
      const int y = i + dy - 1, x = j + dx - 1;
      const float v = ((unsigned)y < IMG_H && (unsigned)x < IMG_W)
                          ? st[(size_t)y * IMG_W + x] : 0.0f;
      sum += hw[dy * 3 + dx] * v;
    }
  const bool weak = (s == 0.5f);
  const bool weak_is_high = (sum > 1.0f) && weak;
  out[idx] = (s == 1.0f ? 1.0f : 0.0f) + (weak_is_high ? 1.0f : 0.0f);
}

// -------------------------------------------------------------------------
extern "C" void kernel_launch(void* const* d_in, const int* in_sizes, int n_in,
                              void* d_out, int out_size, void* d_ws, size_t ws_size,
                              hipStream_t stream) {
  (void)in_sizes; (void)n_in; (void)out_size; (void)ws_size;
  const float* img = (const float*)d_in[0];
  const float* gw  = (const float*)d_in[1];
  const float* sxw = (const float*)d_in[2];
  const float* syw = (const float*)d_in[3];
  // d_in[4] (directional_w) not needed: kernels reduce to neighbor compares.
  const float* hw  = (const float*)d_in[5];

  float* out = (float*)d_out;
  const size_t nBlur = (size_t)NB * NC * PH * PW;
  const size_t nPix  = (size_t)NB * IMG_H * IMG_W;
  float* o_blur = out;
  float* o_gx   = o_blur + nBlur;
  float* o_gy   = o_gx + nPix;
  float* o_mag  = o_gy + nPix;
  float* o_ori  = o_mag + nPix;
  float* o_thin = o_ori + nPix;

  float* ws_summed = (float*)d_ws;                       // NB*PH*PW floats
  float* ws_stage  = ws_summed + (size_t)NB * PH * PW;   // NB*H*W floats

  dim3 g1((PW + 15) / 16, (PH + 15) / 16, NB);
  canny_blur_sum_kernel<<<g1, 32, 0, stream>>>(img, gw, o_blur, ws_summed);

  const int blocks = (int)((nPix + 255) / 256);
  canny_sobel_kernel<<<blocks, 256, 0, stream>>>(ws_summed, sxw, syw,
                                                 o_gx, o_gy, o_mag, o_ori);
  canny_nms_kernel<<<blocks, 256, 0, stream>>>(o_mag, o_ori, ws_stage);
  canny_hyst_kernel<<<blocks, 256, 0, stream>>>(ws_stage, hw, o_thin);
}